// CwRNN_84971632984686
// MI455X (gfx1250) — compile-verified
//
#include <hip/hip_runtime.h>

// Clockwork RNN for MI455X (gfx1250), fp32 WMMA path, round 4.
// Single persistent 32-wave workgroup; h state LDS-resident (263KB of 320KB).
// vs round 3:
//  - merged phases: y-GEMM(t) and candidate-GEMM(t+1) both read the committed
//    h_t, so they share one read-only phase -> 2 barriers/step (was 3) and
//    waves idle during small-Ract candidate steps now overlap the y work.
//  - unchanged: fp32 WMMA 16x16x4, paired row tiles sharing the B operand,
//    distance-2 software-pipelined K loops, block-triangular K-start,
//    active-prefix module skipping, non-temporal output stores.

typedef __attribute__((ext_vector_type(2))) float v2f;
typedef __attribute__((ext_vector_type(8))) float v8f;

#define I_SZ 256
#define H_SZ 1024
#define B_SZ 64
#define T_SZ 2048
#define MS   128          // module size
#define HP   (H_SZ + 4)   // padded LDS pitch (floats): conflict-free r/w

#define WMMA4(a, b, c) \
    __builtin_amdgcn_wmma_f32_16x16x4_f32(false, (a), false, (b), (short)0, (c), false, false)

// D = A(16x4) * B(4x16) + D over K in [kbeg, kend), two row tiles sharing B.
// Distance-2 software pipeline: while the WMMA pair for slice k runs, loads
// for slices k+4 and k+8 are outstanding. Requires (kend-kbeg) >= 8, %4 == 0.
__device__ __forceinline__ void gemm_pair(v8f& acc0, v8f& acc1,
                                          const float* a0p, const float* a1p,
                                          const float* bp,
                                          int kbeg, int kend, int koff)
{
    int k = kbeg + koff;
    v2f b0  = *(const v2f*)(bp  + k);
    v2f a00 = *(const v2f*)(a0p + k);
    v2f a10 = *(const v2f*)(a1p + k);
    k += 4;
    v2f b1  = *(const v2f*)(bp  + k);
    v2f a01 = *(const v2f*)(a0p + k);
    v2f a11 = *(const v2f*)(a1p + k);

    for (int k0 = kbeg; k0 < kend - 8; k0 += 4) {
        const int kn = k0 + 8 + koff;
        v2f b2  = *(const v2f*)(bp  + kn);
        v2f a02 = *(const v2f*)(a0p + kn);
        v2f a12 = *(const v2f*)(a1p + kn);
        acc0 = WMMA4(a00, b0, acc0);
        acc1 = WMMA4(a10, b0, acc1);
        b0 = b1; a00 = a01; a10 = a11;
        b1 = b2; a01 = a02; a11 = a12;
    }
    acc0 = WMMA4(a00, b0, acc0);
    acc1 = WMMA4(a10, b0, acc1);
    acc0 = WMMA4(a01, b1, acc0);
    acc1 = WMMA4(a11, b1, acc1);
}

__device__ __forceinline__ int active_pairs(int t) {
    int nact = (t == 0) ? 8 : __ffs(t);
    if (nact > 8) nact = 8;
    return (nact * MS) >> 3;   // (Ract/16 coltiles) * 2 rowpairs
}

__launch_bounds__(1024, 1)
__global__ void cwrnn_wmma_kernel(const float* __restrict__ x,
                                  const float* __restrict__ w_ih,
                                  const float* __restrict__ w_hh,
                                  const float* __restrict__ b_ih,
                                  const float* __restrict__ b_hh,
                                  const float* __restrict__ fc_w,
                                  const float* __restrict__ fc_b,
                                  float* __restrict__ out)
{
    // 64 * 1028 * 4 = 263,168 bytes of LDS (<= 320KB per WGP on CDNA5)
    __shared__ __align__(16) float h[B_SZ * HP];

    const int tid  = threadIdx.x;
    const int lane = tid & 31;
    const int wave = tid >> 5;     // 0..31
    const int lrow = lane & 15;    // M (A) / N (B,C,D) within a 16x16 tile
    const int lhi  = lane >> 4;    // K-pair select: {0,1} vs {2,3}
    const int koff = lhi * 2;

    for (int i = tid; i < B_SZ * HP; i += 1024) h[i] = 0.0f;
    __syncthreads();

    // candidate-step compute for timestep tc (reads h, x; results -> accs)
    v8f accs[8];
    auto compute_cand = [&](int tc, int npairs) {
        #pragma unroll
        for (int i = 0; i < 4; ++i) {
            const int p = wave + (i << 5);
            if (p < npairs) {
                const int ct      = p >> 1;
                const int rp      = p & 1;              // rows rp*32 .. rp*32+31
                const int colbase = ct << 4;
                const int col     = colbase + lrow;
                const int r0      = (rp << 5) + lrow;   // A row, tile 0 (tile 1 = +16)

                v8f acc0 = {}, acc1 = {};

                // input projection: K = 0..255
                const float* x0 = x + ((size_t)r0 * T_SZ + tc) * I_SZ;
                const float* x1 = x0 + (size_t)16 * T_SZ * I_SZ;
                const float* wi = w_ih + (size_t)col * I_SZ;
                gemm_pair(acc0, acc1, x0, x1, wi, 0, I_SZ, koff);

                // recurrent: block-triangular -> K in [128*(colbase/128), 1024)
                const int kstart = (colbase / MS) * MS;
                const float* h0 = &h[(size_t)r0 * HP];
                const float* h1 = h0 + (size_t)16 * HP;
                const float* wh = w_hh + (size_t)col * H_SZ;
                gemm_pair(acc0, acc1, h0, h1, wh, kstart, H_SZ, koff);

                const float bias = b_ih[col] + b_hh[col];
                #pragma unroll
                for (int v = 0; v < 8; ++v) {
                    acc0[v] = tanhf(acc0[v] + bias);
                    acc1[v] = tanhf(acc1[v] + bias);
                }
                accs[2 * i]     = acc0;
                accs[2 * i + 1] = acc1;
            }
        }
    };

    // prologue: candidates for t = 0 (h == 0)
    compute_cand(0, active_pairs(0));

    for (int t = 0; t < T_SZ; ++t) {
        const int npairs = active_pairs(t);

        __syncthreads();   // all reads of h_{t-1} (cand) / h_t-producing state done

        // ---- commit active columns: h becomes h_t ----
        #pragma unroll
        for (int i = 0; i < 4; ++i) {
            const int p = wave + (i << 5);
            if (p < npairs) {
                const int ct  = p >> 1;
                const int rp  = p & 1;
                const int col = (ct << 4) + lrow;
                const int rb  = (rp << 5) + 8 * lhi;   // C/D layout: M = v (+8 hi half)
                v8f acc0 = accs[2 * i];
                v8f acc1 = accs[2 * i + 1];
                #pragma unroll
                for (int v = 0; v < 8; ++v) {
                    h[(size_t)(rb + v) * HP + col]      = acc0[v];
                    h[(size_t)(rb + 16 + v) * HP + col] = acc1[v];
                }
            }
        }
        __syncthreads();   // h_t visible to everyone

        // ---- merged read-only phase on h_t: y(t) and cand(t+1) ----
        // y_t = h_t @ fc_w^T + fc_b   (64x256 -> exactly 1 pair/wave)
        {
            const int ct      = wave >> 1;
            const int rp      = wave & 1;
            const int colbase = ct << 4;
            const int col     = colbase + lrow;
            const int r0      = (rp << 5) + lrow;

            v8f acc0 = {}, acc1 = {};
            const float* h0 = &h[(size_t)r0 * HP];
            const float* h1 = h0 + (size_t)16 * HP;
            const float* wo = fc_w + (size_t)col * H_SZ;
            gemm_pair(acc0, acc1, h0, h1, wo, 0, H_SZ, koff);

            const float bias = fc_b[col];
            const int rb = (rp << 5) + 8 * lhi;
            #pragma unroll
            for (int v = 0; v < 8; ++v) {
                // write-once output stream: non-temporal, don't pollute L2
                __builtin_nontemporal_store(acc0[v] + bias,
                    &out[((size_t)(rb + v) * T_SZ + t) * I_SZ + col]);
                __builtin_nontemporal_store(acc1[v] + bias,
                    &out[((size_t)(rb + 16 + v) * T_SZ + t) * I_SZ + col]);
            }
        }
        // candidates for t+1 (also reads only h_t; committed next iteration)
        if (t + 1 < T_SZ)
            compute_cand(t + 1, active_pairs(t + 1));
    }
}

extern "C" void kernel_launch(void* const* d_in, const int* in_sizes, int n_in,
                              void* d_out, int out_size, void* d_ws, size_t ws_size,
                              hipStream_t stream) {
    (void)in_sizes; (void)n_in; (void)out_size; (void)d_ws; (void)ws_size;
    const float* x   = (const float*)d_in[0];
    const float* wih = (const float*)d_in[1];
    const float* whh = (const float*)d_in[2];
    const float* bih = (const float*)d_in[3];
    const float* bhh = (const float*)d_in[4];
    const float* fcw = (const float*)d_in[5];
    const float* fcb = (const float*)d_in[6];
    // d_in[7] = n_modules (==8, hardcoded)
    float* out = (float*)d_out;

    hipLaunchKernelGGL(cwrnn_wmma_kernel, dim3(1), dim3(1024), 0, stream,
                       x, wih, whh, bih, bhh, fcw, fcb, out);
}